// PlaneEmbeddingNetwork_10402410791494
// MI455X (gfx1250) — compile-verified
//
#include <hip/hip_runtime.h>
#include <hip/hip_bf16.h>

#define FACES 300000
#define S 8
#define E 16
#define HID 32
#define OUT 32

typedef __attribute__((ext_vector_type(2))) float v2f;
typedef __attribute__((ext_vector_type(8))) float v8f;

__device__ __forceinline__ v8f wmma_f32(v2f a, v2f b, v8f c) {
  // D = A(16x4,f32) * B(4x16,f32) + C(16x16,f32)
  return __builtin_amdgcn_wmma_f32_16x16x4_f32(
      /*neg_a=*/false, a, /*neg_b=*/false, b,
      /*c_mod=*/(short)0, c, /*reuse_a=*/false, /*reuse_b=*/false);
}

// ---- DPP8 permutes within 8-lane groups (VALU, no LDS pipeline) ----
constexpr int dpp8_sel(int a, int b, int c, int d, int e, int f, int g, int h) {
  return a | (b << 3) | (c << 6) | (d << 9) | (e << 12) | (f << 15) | (g << 18) | (h << 21);
}
constexpr int SEL_X1 = dpp8_sel(1, 0, 3, 2, 5, 4, 7, 6);  // lane i reads i^1
constexpr int SEL_X2 = dpp8_sel(2, 3, 0, 1, 6, 7, 4, 5);  // lane i reads i^2
constexpr int SEL_X4 = dpp8_sel(4, 5, 6, 7, 0, 1, 2, 3);  // lane i reads i^4

template <int SEL>
__device__ __forceinline__ float dpp8_perm(float x) {
  return __int_as_float(__builtin_amdgcn_mov_dpp8(__float_as_int(x), SEL));
}
template <int J>
__device__ __forceinline__ float dpp8_bcast(float x) {   // broadcast group-lane J
  return dpp8_perm<dpp8_sel(J, J, J, J, J, J, J, J)>(x);
}
__device__ __forceinline__ float group8_max(float x) {
  x = fmaxf(x, dpp8_perm<SEL_X1>(x));
  x = fmaxf(x, dpp8_perm<SEL_X2>(x));
  x = fmaxf(x, dpp8_perm<SEL_X4>(x));
  return x;
}
__device__ __forceinline__ float group8_sum(float x) {
  x += dpp8_perm<SEL_X1>(x);
  x += dpp8_perm<SEL_X2>(x);
  x += dpp8_perm<SEL_X4>(x);
  return x;
}

// Reduce-scatter over an 8-lane group: input P[j] = per-lane partial of dot j.
// Output: lane at group position g returns full sum of P[g] over the 8 lanes.
__device__ __forceinline__ float rscatter8(const float P[8], bool s4, bool s2, bool s1) {
  float q[4];
  #pragma unroll
  for (int j = 0; j < 4; ++j) {
    float mine = s4 ? P[j + 4] : P[j];
    float give = s4 ? P[j] : P[j + 4];
    q[j] = mine + dpp8_perm<SEL_X4>(give);   // covers lanes {g, g^4}, index j+(g&4)
  }
  float r[2];
  #pragma unroll
  for (int j = 0; j < 2; ++j) {
    float mine = s2 ? q[j + 2] : q[j];
    float give = s2 ? q[j] : q[j + 2];
    r[j] = mine + dpp8_perm<SEL_X2>(give);   // index j+(g&2)+(g&4), 4 lanes
  }
  float mine = s1 ? r[1] : r[0];
  float give = s1 ? r[0] : r[1];
  return mine + dpp8_perm<SEL_X1>(give);     // index g, all 8 lanes
}

__global__ __launch_bounds__(256) void plane_embed_fused(
    const int* __restrict__ face_indices,   // [F,S]
    const float* __restrict__ node_embed,   // [N,E]
    const float* __restrict__ in_proj_w,    // [3E,E]
    const float* __restrict__ in_proj_b,    // [3E]
    const float* __restrict__ out_proj_w,   // [E,E]
    const float* __restrict__ out_proj_b,   // [E]
    const float* __restrict__ fc_w,         // [HID,E]
    const float* __restrict__ fc_b,         // [HID]
    const float* __restrict__ fc2_w,        // [OUT,HID]
    const float* __restrict__ fc2_b,        // [OUT]
    float* __restrict__ out)                // [F,OUT]
{
  __shared__ float lds_all[8][16 * 33];     // per-wave 16x32 staging tile (stride 33)
  const int lane  = threadIdx.x & 31;
  const int wslot = threadIdx.x >> 5;
  const int wave  = blockIdx.x * 8 + wslot;
  if (wave >= FACES / 16) return;           // wave-uniform: EXEC stays all-ones
  const long gface = (long)wave * 16;       // first of 16 faces handled by this wave
  float* tile = lds_all[wslot];

  const int r    = lane & 15;               // A-row fed / C-column owned by this lane
  const int half = lane >> 4;               // 0 -> K{0,1}, 1 -> K{2,3} within a k=4 step
  const int koff = half * 2;
  const bool s4 = (lane & 4) != 0;          // group-position bits within 8-lane group
  const bool s2 = (lane & 2) != 0;
  const bool s1 = (lane & 1) != 0;

  float bq[3];
  #pragma unroll
  for (int t = 0; t < 3; ++t) bq[t] = in_proj_b[t * 16 + r];

  // softmax exponent scale: log2(e) / sqrt(DH), DH = 8
  const float expsc = 1.4426950408889634f * 0.35355339059327373f;

  // ---------------- per-pair attention: 8 iterations x (gather + 12 WMMA + VALU attn)
  for (int p = 0; p < 8; ++p) {
    const int fpair = (int)gface + 2 * p;
    const int faceA = fpair + (r >> 3);     // rows 0-7 -> face fpair, 8-15 -> fpair+1
    const int strk  = r & 7;
    const int idx   = face_indices[faceA * S + strk];
    const float* src = node_embed + (long)idx * E;

    v2f a[4];
    #pragma unroll
    for (int i = 0; i < 4; ++i)
      a[i] = *(const v2f*)(src + 4 * i + koff);

    // qkv = strokes @ W_in^T + b : three 16-col tiles, K=16 as 4 chained k=4 WMMAs
    v8f qkv[3];
    #pragma unroll
    for (int t = 0; t < 3; ++t) {
      v8f ct;
      #pragma unroll
      for (int j = 0; j < 8; ++j) ct[j] = bq[t];
      #pragma unroll
      for (int i = 0; i < 4; ++i) {
        v2f b = *(const v2f*)(in_proj_w + (t * 16 + r) * E + 4 * i + koff);
        ct = wmma_f32(a[i], b, ct);
      }
      qkv[t] = ct;
    }

    // attention: lane owns channel r of face (fpair+half); head = r>>3; group pos g=lane&7.
    // mean-over-queries commutes with attn@V: mean_c = sum_j w_j * v[j][c],
    // w_j = (1/8) * sum_i softmax_j(s[i][:]). Reduce-scatter leaves s[i][g] at lane g,
    // so each lane does ONE exp per query row instead of eight.
    float w = 0.f;
    #pragma unroll
    for (int i = 0; i < 8; ++i) {
      float P[8];
      #pragma unroll
      for (int j = 0; j < 8; ++j) P[j] = qkv[0][i] * qkv[1][j];
      float s  = rscatter8(P, s4, s2, s1);            // raw dot, scale folded into exp
      float mx = group8_max(s);
      float e  = __builtin_amdgcn_exp2f((s - mx) * expsc);   // v_exp_f32 (1 per lane)
      float den = group8_sum(e);
      w += e * __builtin_amdgcn_rcpf(den);            // p[i][g], accumulated over i
    }
    // mean_c = (1/8) * sum_j w_j * v[j][c]  (broadcast w_j from group-lane j)
    float m;
    m  = dpp8_bcast<0>(w) * qkv[2][0];
    m += dpp8_bcast<1>(w) * qkv[2][1];
    m += dpp8_bcast<2>(w) * qkv[2][2];
    m += dpp8_bcast<3>(w) * qkv[2][3];
    m += dpp8_bcast<4>(w) * qkv[2][4];
    m += dpp8_bcast<5>(w) * qkv[2][5];
    m += dpp8_bcast<6>(w) * qkv[2][6];
    m += dpp8_bcast<7>(w) * qkv[2][7];
    m *= 0.125f;                              // mean over S (commutes with out_proj)
    tile[(2 * p + half) * 33 + r] = m;        // row = face-in-group, col = channel
  }
  asm volatile("s_wait_dscnt 0" ::: "memory");

  // ---------------- out_proj on batched face means: [16 faces x 16] @ Wo^T + bo
  v2f a2[4];
  #pragma unroll
  for (int i = 0; i < 4; ++i) {
    int k = 4 * i + koff;
    v2f a; a.x = tile[r * 33 + k]; a.y = tile[r * 33 + k + 1];
    a2[i] = a;
  }
  v8f d1;
  { float bias = out_proj_b[r];
    #pragma unroll
    for (int j = 0; j < 8; ++j) d1[j] = bias; }
  #pragma unroll
  for (int i = 0; i < 4; ++i) {
    v2f b = *(const v2f*)(out_proj_w + r * E + 4 * i + koff);
    d1 = wmma_f32(a2[i], b, d1);
  }
  asm volatile("s_wait_dscnt 0" ::: "memory");  // WAR: means fully read before overwrite
  #pragma unroll
  for (int j = 0; j < 8; ++j) tile[(j + half * 8) * 33 + r] = d1[j];
  asm volatile("s_wait_dscnt 0" ::: "memory");

  // ---------------- fc: [16 x 16] @ fc_w^T + b, ReLU ; HID=32 -> two 16-col tiles
  v2f a3[4];
  #pragma unroll
  for (int i = 0; i < 4; ++i) {
    int k = 4 * i + koff;
    v2f a; a.x = tile[r * 33 + k]; a.y = tile[r * 33 + k + 1];
    a3[i] = a;
  }
  v8f d2[2];
  #pragma unroll
  for (int b2 = 0; b2 < 2; ++b2) {
    int n = b2 * 16 + r;
    float bias = fc_b[n];
    v8f ct;
    #pragma unroll
    for (int j = 0; j < 8; ++j) ct[j] = bias;
    #pragma unroll
    for (int i = 0; i < 4; ++i) {
      v2f b = *(const v2f*)(fc_w + n * E + 4 * i + koff);
      ct = wmma_f32(a3[i], b, ct);
    }
    #pragma unroll
    for (int j = 0; j < 8; ++j) ct[j] = fmaxf(ct[j], 0.f);
    d2[b2] = ct;
  }
  asm volatile("s_wait_dscnt 0" ::: "memory");
  #pragma unroll
  for (int b2 = 0; b2 < 2; ++b2)
    #pragma unroll
    for (int j = 0; j < 8; ++j)
      tile[(j + half * 8) * 33 + b2 * 16 + r] = d2[b2][j];
  asm volatile("s_wait_dscnt 0" ::: "memory");

  // ---------------- fc2: [16 x 32] @ fc2_w^T + b ; K=32 -> 8 chained steps per tile
  v2f a4[8];
  #pragma unroll
  for (int i = 0; i < 8; ++i) {
    int k = 4 * i + koff;
    v2f a; a.x = tile[r * 33 + k]; a.y = tile[r * 33 + k + 1];
    a4[i] = a;
  }
  #pragma unroll
  for (int b3 = 0; b3 < 2; ++b3) {
    int n = b3 * 16 + r;
    float bias = fc2_b[n];
    v8f ct;
    #pragma unroll
    for (int j = 0; j < 8; ++j) ct[j] = bias;
    #pragma unroll
    for (int i = 0; i < 8; ++i) {
      v2f b = *(const v2f*)(fc2_w + n * HID + 4 * i + koff);
      ct = wmma_f32(a4[i], b, ct);
    }
    #pragma unroll
    for (int j = 0; j < 8; ++j) {
      long face = gface + j + half * 8;
      out[face * OUT + n] = ct[j];
    }
  }
}

extern "C" void kernel_launch(void* const* d_in, const int* in_sizes, int n_in,
                              void* d_out, int out_size, void* d_ws, size_t ws_size,
                              hipStream_t stream) {
  const int*   face_indices = (const int*)  d_in[0];
  const float* node_embed   = (const float*)d_in[1];
  const float* in_proj_w    = (const float*)d_in[2];
  const float* in_proj_b    = (const float*)d_in[3];
  const float* out_proj_w   = (const float*)d_in[4];
  const float* out_proj_b   = (const float*)d_in[5];
  const float* fc_w         = (const float*)d_in[6];
  const float* fc_b         = (const float*)d_in[7];
  const float* fc2_w        = (const float*)d_in[8];
  const float* fc2_b        = (const float*)d_in[9];
  float* out = (float*)d_out;

  const int waves   = FACES / 16;                 // 18750 waves, 16 faces each
  const int threads = 256;                        // 8 waves per block
  const int blocks  = (waves * 32 + threads - 1) / threads;
  plane_embed_fused<<<blocks, threads, 0, stream>>>(
      face_indices, node_embed, in_proj_w, in_proj_b,
      out_proj_w, out_proj_b, fc_w, fc_b, fc2_w, fc2_b, out);
}